// LCCLoss_49409303773369
// MI455X (gfx1250) — compile-verified
//
#include <hip/hip_runtime.h>

// ---------------- problem constants (from reference) ----------------
#define HDIM 512
#define WDIM 512
#define WINR 9          // window
#define RAD  4          // window/2
#define EPSV 1e-5f

// ---------------- tiling ----------------
#define NT    64                 // threads per block (2 wave32)
#define CPT   4                  // columns per thread
#define TW    (NT * CPT)         // 256 columns per workgroup
#define NSTRIP (WDIM / TW)       // 2
#define BH    128                // band height (rows per workgroup)
#define NBAND (HDIM / BH)        // 4
#define EPS81 (EPSV * 81.0f * 81.0f)

typedef __attribute__((ext_vector_type(2))) float v2f;
typedef __attribute__((ext_vector_type(8))) float v8f;

// ---------------- CDNA5 async global->LDS staging (guarded) ----------------
#ifndef __has_builtin
#define __has_builtin(x) 0
#endif

#if __has_builtin(__builtin_amdgcn_global_load_async_to_lds_b128)
#define USE_ASYNC_LDS 1
// Builtin signature (from hipcc diagnostic): pointers to int vector_size(16)
// with global (AS1) source and LDS (AS3) destination, plus imm offset / cpol.
typedef int v4i_gcc __attribute__((vector_size(16)));
typedef __attribute__((address_space(1))) v4i_gcc glob_v4i;
typedef __attribute__((address_space(3))) v4i_gcc lds_v4i;
static __device__ __forceinline__ void async_copy16(const float* g, float* l) {
  // GLOBAL_LOAD_ASYNC_TO_LDS_B128: memory -> LDS, tracked by ASYNCcnt
  __builtin_amdgcn_global_load_async_to_lds_b128((glob_v4i*)g, (lds_v4i*)l, 0, 0);
}
static __device__ __forceinline__ void async_wait0() {
#if __has_builtin(__builtin_amdgcn_s_wait_asynccnt)
  __builtin_amdgcn_s_wait_asynccnt(0);
#else
  asm volatile("s_wait_asynccnt 0x0" ::: "memory");
#endif
}
#else
#define USE_ASYNC_LDS 0
#endif

// =====================================================================
// Pass 1: fused separable 9x9 box sums + cc, one partial sum per WG.
// =====================================================================
__global__ __launch_bounds__(NT) void lcc_main(const float* __restrict__ A,
                                               const float* __restrict__ Bm,
                                               float* __restrict__ partial) {
  // double-buffered row stage: [parity][img][x], x covers [x0-4, x0+TW+4)
  __shared__ float stage[2][2][TW + 2 * RAD];
  // 9-row ring of horizontal window sums (thread-private columns -> no barrier)
  __shared__ float ringA[WINR][TW];
  __shared__ float ringB[WINR][TW];
  __shared__ float ringAA[WINR][TW];
  __shared__ float ringBB[WINR][TW];
  __shared__ float ringAB[WINR][TW];
  __shared__ float red[NT];

  const int tid   = threadIdx.x;
  const int strip = blockIdx.x & (NSTRIP - 1);
  const int band  = blockIdx.x / NSTRIP;
  const int b     = blockIdx.y;
  const int x0    = strip * TW;
  const int y0    = band * BH;
  const int c0    = tid * CPT;   // local column base

  const float* __restrict__ aimg = A  + (size_t)b * HDIM * WDIM;
  const float* __restrict__ bimg = Bm + (size_t)b * HDIM * WDIM;

  // Zero the 8 halo float4 slots once (2 parities x 2 imgs x 2 sides).
  // In-loop, halos are only overwritten when the source columns are in-bounds,
  // so out-of-image halos remain zero for the whole band.
  if (tid < 8) {
    const int par  = tid & 1;
    const int img  = (tid >> 1) & 1;
    const int side = (tid >> 2) & 1;
    *reinterpret_cast<float4*>(&stage[par][img][side ? (RAD + TW) : 0]) =
        make_float4(0.f, 0.f, 0.f, 0.f);
  }

  float vsa[CPT], vsb[CPT], vsaa[CPT], vsbb[CPT], vsab[CPT];
#pragma unroll
  for (int c = 0; c < CPT; ++c) { vsa[c] = vsb[c] = vsaa[c] = vsbb[c] = vsab[c] = 0.f; }

  float acc = 0.f;

  // Unified prime + main loop: staged row r = y0-5 .. y0+BH+3.
  // After processing row r, vs* = sum of h over rows [r-8, r] == window of y=r-4.
  for (int it = 0; it < BH + WINR; ++it) {
    const int  r        = y0 - 5 + it;
    const bool rowvalid = (r >= 0) && (r < HDIM);
    const bool warm     = (it >= WINR);                 // ring holds h(r-9); output y=r-4 valid
    const int  p        = (r + 16) & 1;                 // stage parity (r >= -5)
    const int  slot     = ((r % WINR) + WINR) % WINR;   // ring slot, safe for negative r

    // ---- stage row r into LDS (vectorized, with 4-col halos) ----
    if (rowvalid) {
      const float* rowA = aimg + (size_t)r * WDIM + x0;
      const float* rowB = bimg + (size_t)r * WDIM + x0;
#if USE_ASYNC_LDS
      async_copy16(rowA + 4 * tid, &stage[p][0][RAD + c0]);
      async_copy16(rowB + 4 * tid, &stage[p][1][RAD + c0]);
      if (tid == 0 && x0 > 0) {
        async_copy16(rowA - RAD, &stage[p][0][0]);
        async_copy16(rowB - RAD, &stage[p][1][0]);
      }
      if (tid == NT - 1 && x0 + TW < WDIM) {
        async_copy16(rowA + TW, &stage[p][0][RAD + TW]);
        async_copy16(rowB + TW, &stage[p][1][RAD + TW]);
      }
      async_wait0();
#else
      const float4* ra = reinterpret_cast<const float4*>(rowA);
      const float4* rb = reinterpret_cast<const float4*>(rowB);
      reinterpret_cast<float4*>(&stage[p][0][RAD])[tid] = ra[tid];
      reinterpret_cast<float4*>(&stage[p][1][RAD])[tid] = rb[tid];
      if (tid == 0 && x0 > 0) {
        *reinterpret_cast<float4*>(&stage[p][0][0]) =
            *reinterpret_cast<const float4*>(rowA - RAD);
        *reinterpret_cast<float4*>(&stage[p][1][0]) =
            *reinterpret_cast<const float4*>(rowB - RAD);
      }
      if (tid == NT - 1 && x0 + TW < WDIM) {
        *reinterpret_cast<float4*>(&stage[p][0][RAD + TW]) =
            *reinterpret_cast<const float4*>(rowA + TW);
        *reinterpret_cast<float4*>(&stage[p][1][RAD + TW]) =
            *reinterpret_cast<const float4*>(rowB + TW);
      }
#endif
      // warm L2 two rows ahead (global_prefetch_b8)
      const int rp = r + 2;
      if (rp < HDIM) {
        __builtin_prefetch(aimg + (size_t)rp * WDIM + x0 + c0, 0, 0);
        __builtin_prefetch(bimg + (size_t)rp * WDIM + x0 + c0, 0, 0);
      }
    }
    __syncthreads();

    // ---- horizontal 9-tap sums for this thread's 4 columns (sliding) ----
    float av[12] = {}, bv[12] = {};
    if (rowvalid) {
      const float4* s4a = reinterpret_cast<const float4*>(&stage[p][0][c0]);
      const float4* s4b = reinterpret_cast<const float4*>(&stage[p][1][c0]);
      float4 a0 = s4a[0], a1 = s4a[1], a2 = s4a[2];
      float4 b0 = s4b[0], b1 = s4b[1], b2 = s4b[2];
      av[0]=a0.x; av[1]=a0.y; av[2]=a0.z; av[3]=a0.w;
      av[4]=a1.x; av[5]=a1.y; av[6]=a1.z; av[7]=a1.w;
      av[8]=a2.x; av[9]=a2.y; av[10]=a2.z; av[11]=a2.w;
      bv[0]=b0.x; bv[1]=b0.y; bv[2]=b0.z; bv[3]=b0.w;
      bv[4]=b1.x; bv[5]=b1.y; bv[6]=b1.z; bv[7]=b1.w;
      bv[8]=b2.x; bv[9]=b2.y; bv[10]=b2.z; bv[11]=b2.w;
    }

    float sa = 0.f, sb = 0.f, saa = 0.f, sbb = 0.f, sab = 0.f;
    if (rowvalid) {
#pragma unroll
      for (int t = 0; t < WINR; ++t) {
        const float a_ = av[t], b_ = bv[t];
        sa += a_; sb += b_;
        saa = fmaf(a_, a_, saa);
        sbb = fmaf(b_, b_, sbb);
        sab = fmaf(a_, b_, sab);
      }
    }

#pragma unroll
    for (int c = 0; c < CPT; ++c) {
      if (rowvalid && c > 0) {
        const float ai = av[c + 8], ao = av[c - 1];
        const float bi = bv[c + 8], bo = bv[c - 1];
        sa  += ai - ao;           sb  += bi - bo;
        saa += ai * ai - ao * ao; sbb += bi * bi - bo * bo;
        sab += ai * bi - ao * bo;
      }
      const int lx = c0 + c;
      // ring update + vertical running sums (h of OOB rows is 0; sa..sab stay 0 then)
      const float oA  = warm ? ringA [slot][lx] : 0.f;
      const float oB  = warm ? ringB [slot][lx] : 0.f;
      const float oAA = warm ? ringAA[slot][lx] : 0.f;
      const float oBB = warm ? ringBB[slot][lx] : 0.f;
      const float oAB = warm ? ringAB[slot][lx] : 0.f;
      vsa [c] += sa  - oA;
      vsb [c] += sb  - oB;
      vsaa[c] += saa - oAA;
      vsbb[c] += sbb - oBB;
      vsab[c] += sab - oAB;
      ringA [slot][lx] = sa;
      ringB [slot][lx] = sb;
      ringAA[slot][lx] = saa;
      ringBB[slot][lx] = sbb;
      ringAB[slot][lx] = sab;

      if (warm) {  // output row y = r-4 in [y0, y0+BH)
        const float Sa = vsa[c], Sb = vsb[c];
        const float va  = fmaf(81.f, vsaa[c], -Sa * Sa);
        const float vb_ = fmaf(81.f, vsbb[c], -Sb * Sb);
        const float num = fmaf(81.f, vsab[c], -Sa * Sb);
        const float den = sqrtf(va * vb_) + EPS81;
        acc += num / den;
      }
    }
  }

  // ---- workgroup reduction (deterministic) ----
  red[tid] = acc;
  __syncthreads();
#pragma unroll
  for (int s = NT / 2; s > 0; s >>= 1) {
    if (tid < s) red[tid] += red[tid + s];
    __syncthreads();
  }
  if (tid == 0) partial[blockIdx.y * (NSTRIP * NBAND) + blockIdx.x] = red[0];
}

// =====================================================================
// Pass 2: single-wave deterministic reduction of partials via
// chained V_WMMA_F32_16X16X4_F32 (B = ones), then 1 - mean.
// =====================================================================
__global__ __launch_bounds__(32) void lcc_reduce(const float* __restrict__ partial,
                                                 float* __restrict__ out,
                                                 int n, float invN) {
  const int lane = threadIdx.x;
  v8f c = {0.f, 0.f, 0.f, 0.f, 0.f, 0.f, 0.f, 0.f};
  v2f ones; ones[0] = 1.f; ones[1] = 1.f;

  const int nb = n & ~63;                     // 64 values per WMMA (16x4 A-matrix)
  for (int base = 0; base < nb; base += 64) {
    v2f a;
    a[0] = partial[base + lane];
    a[1] = partial[base + 32 + lane];
    // D[m][n] = sum_k A[m][k] + C[m][n]  (B all-ones) -> accumulate chunk row-sums
    c = __builtin_amdgcn_wmma_f32_16x16x4_f32(false, a, false, ones,
                                              (short)0, c, false, false);
  }
  // total = sum_m C[m][0]: lanes 0/16 hold N=0 for M=vgpr / M=vgpr+8
  float s = c[0] + c[1] + c[2] + c[3] + c[4] + c[5] + c[6] + c[7];
  float tot = __shfl(s, 0, 32) + __shfl(s, 16, 32);

  float tail = 0.f;
  for (int i = nb + lane; i < n; i += 32) tail += partial[i];
#pragma unroll
  for (int off = 16; off > 0; off >>= 1) tail += __shfl_down(tail, off, 32);

  if (lane == 0) out[0] = 1.0f - (tot + tail) * invN;
}

// =====================================================================
extern "C" void kernel_launch(void* const* d_in, const int* in_sizes, int n_in,
                              void* d_out, int out_size, void* d_ws, size_t ws_size,
                              hipStream_t stream) {
  const float* a = (const float*)d_in[0];
  const float* b = (const float*)d_in[1];
  float* out     = (float*)d_out;
  float* partial = (float*)d_ws;

  const int Bc = in_sizes[0] / (HDIM * WDIM);   // 64
  dim3 grid(NSTRIP * NBAND, Bc);                // 8 x 64 = 512 workgroups
  hipLaunchKernelGGL(lcc_main, grid, dim3(NT), 0, stream, a, b, partial);

  const int nwg = Bc * NSTRIP * NBAND;          // 512 partials
  const float invN = 1.0f / ((float)Bc * (float)HDIM * (float)WDIM);
  hipLaunchKernelGGL(lcc_reduce, dim3(1), dim3(32), 0, stream, partial, out, nwg, invN);
}